// CondAttLSTM_58548994179239
// MI455X (gfx1250) — compile-verified
//
#include <hip/hip_runtime.h>

// ---------------------------------------------------------------------------
// Problem constants (from the reference)
// ---------------------------------------------------------------------------
constexpr int T_STEPS = 512;   // decode steps
constexpr int LCTX    = 256;   // context length
constexpr int DD      = 512;   // D_IN == D_OUT == D_CTX
constexpr int AA      = 256;   // attention hidden dim
constexpr int GG      = 2048;  // 4 * D_OUT

constexpr int NWG  = 32;       // persistent workgroups for the scan
constexpr int NTHR = 256;      // threads per workgroup (8 wave32)

// workspace layout (float offsets). barrier counter lives in bytes [0,256)
constexpr size_t OFF_CTXVEC = 64;                      // [512]
constexpr size_t OFF_HCTX   = OFF_CTXVEC + DD;         // [512]
constexpr size_t OFF_GATES  = OFF_HCTX + DD;           // [2048]
constexpr size_t OFF_CTXT   = 4096;                    // ctxT transposed [A][L]
constexpr size_t OFF_XWX    = OFF_CTXT + (size_t)AA * LCTX;     // [T][G]
constexpr size_t OFF_HWT    = OFF_XWX + (size_t)T_STEPS * GG;   // histWhh^T [A][T]

typedef float v8f __attribute__((ext_vector_type(8)));
typedef float v2f __attribute__((ext_vector_type(2)));

// ---------------------------------------------------------------------------
// Precompute GEMM: C[M,N] = A[M,K] @ B[K,N] + bias[N], fp32 via WMMA 16x16x4.
// One wave per 16x16 output tile. Layouts per CDNA5 ISA 7.12.2:
//   A 16x4 f32: lanes 0-15 rows M=0..15 {K=k,k+1}; lanes 16-31 {K=k+2,k+3}
//   C 16x16 f32: VGPR v -> M=v (lanes 0-15) / M=v+8 (lanes 16-31), N=lane&15
// transposeStore=1 stores C^T (used for ctxT so the scan reads coalesced).
// ---------------------------------------------------------------------------
__global__ void __launch_bounds__(256)
wmma_gemm_f32_kernel(const float* __restrict__ A, const float* __restrict__ B,
                     const float* __restrict__ bias, float* __restrict__ C,
                     int M, int N, int K, int transposeStore) {
  const int lane = threadIdx.x & 31;
  const int wave = blockIdx.x * (blockDim.x >> 5) + (threadIdx.x >> 5);
  const int ntN  = N >> 4;
  const int mt   = wave / ntN;
  const int nt   = wave - mt * ntN;
  if (mt * 16 >= M) return;                // grids are sized exactly; uniform
  const int half = lane >> 4;              // 0: lanes 0-15, 1: lanes 16-31
  const int l16  = lane & 15;
  const int row  = mt * 16 + l16;          // A row owned by this lane
  const int col  = nt * 16 + l16;          // B/C column owned by this lane

  v8f acc = {};
  for (int k = 0; k < K; k += 4) {
    const int ka = k + (half << 1);
    v2f a, b;
    a[0] = A[(size_t)row * K + ka];
    a[1] = A[(size_t)row * K + ka + 1];
    b[0] = B[(size_t)ka * N + col];
    b[1] = B[(size_t)(ka + 1) * N + col];
    acc = __builtin_amdgcn_wmma_f32_16x16x4_f32(false, a, false, b,
                                                (short)0, acc, false, false);
  }
#pragma unroll
  for (int v = 0; v < 8; ++v) {
    const int m = mt * 16 + v + half * 8;
    const float val = acc[v] + bias[col];
    if (transposeStore) C[(size_t)col * M + m] = val;
    else                C[(size_t)m * N + col] = val;
  }
}

// ---------------------------------------------------------------------------
// Monotonic device-wide software barrier (32 resident WGs, persistent kernel)
// ---------------------------------------------------------------------------
__device__ inline void global_barrier(unsigned* ctr, unsigned target) {
  __builtin_amdgcn_fence(__ATOMIC_RELEASE, "agent");  // publish my writes
  __syncthreads();
  if (threadIdx.x == 0) {
    __hip_atomic_fetch_add(ctr, 1u, __ATOMIC_RELAXED, __HIP_MEMORY_SCOPE_AGENT);
    while (__hip_atomic_load(ctr, __ATOMIC_RELAXED, __HIP_MEMORY_SCOPE_AGENT) < target)
      __builtin_amdgcn_s_sleep(2);
  }
  __syncthreads();
  __builtin_amdgcn_fence(__ATOMIC_ACQUIRE, "agent");  // see others' writes
}

__device__ inline float sigmoidf_(float x) { return 1.f / (1.f + __expf(-x)); }

// ---------------------------------------------------------------------------
// Persistent sequential scan: 32 WGs x 256 threads, 3 global barriers / step
// ---------------------------------------------------------------------------
__global__ void __launch_bounds__(NTHR)
scan_kernel(const float* __restrict__ h0,
            const float* __restrict__ Uh,  const float* __restrict__ Cg,
            const float* __restrict__ Pg,  const float* __restrict__ Hg,
            const float* __restrict__ context,
            const float* __restrict__ Wah, const float* __restrict__ wa,
            const float* __restrict__ ba,
            const float* __restrict__ Whh, const float* __restrict__ bhh,
            const float* __restrict__ Wha, const float* __restrict__ wh,
            const float* __restrict__ bh,
            const int* __restrict__ parent_t,
            float* __restrict__ out_h, float* __restrict__ out_ctx,
            float* __restrict__ ws, unsigned* __restrict__ barctr) {
  const int wg  = blockIdx.x;
  const int tid = threadIdx.x;

  float* cvec  = ws + OFF_CTXVEC;
  float* hctx  = ws + OFF_HCTX;
  float* gates = ws + OFF_GATES;
  const float* ctxT = ws + OFF_CTXT;   // [A][L]
  const float* xwx  = ws + OFF_XWX;    // [T][G]
  float* hwt        = ws + OFF_HWT;    // histWhh^T [A][T]

  __shared__ float s_h[DD];            // current hidden state (LDS-resident)
  __shared__ float s_vecA[AA];         // h@Wah or h@Wha
  __shared__ float s_sc[T_STEPS];      // attention scores / weights
  __shared__ float s_red[NTHR];        // reductions
  __shared__ float s_part[NTHR];       // gate partial sums
  __shared__ float s_cv[DD], s_hc[DD]; // staged ctx_vec / h_ctx

  s_h[tid] = h0[tid];
  s_h[tid + 256] = h0[tid + 256];
  __syncthreads();

  unsigned tgt = 0;

  for (int t = 0; t < T_STEPS; ++t) {
    const int  p    = parent_t[t];
    const bool live = (t > 0);

    // ---------------- PHASE A ------------------------------------------
    if (wg == 0) {
      // context attention -> ctx_vec
      float s = 0.f;
      for (int k = 0; k < DD; ++k) s += s_h[k] * Wah[(size_t)k * AA + tid];
      s_vecA[tid] = s;
      __syncthreads();
      float sc = ba[0];
      for (int a = 0; a < AA; ++a)
        sc += wa[a] * tanhf(ctxT[(size_t)a * LCTX + tid] + s_vecA[a]);
      s_red[tid] = sc; __syncthreads();
      for (int st = 128; st > 0; st >>= 1) {
        if (tid < st) s_red[tid] = fmaxf(s_red[tid], s_red[tid + st]);
        __syncthreads();
      }
      const float mx = s_red[0]; __syncthreads();
      const float e  = __expf(sc - mx);
      s_red[tid] = e; __syncthreads();
      for (int st = 128; st > 0; st >>= 1) {
        if (tid < st) s_red[tid] += s_red[tid + st];
        __syncthreads();
      }
      const float inv = 1.f / s_red[0]; __syncthreads();
      s_sc[tid] = e * inv; __syncthreads();
      for (int rep = 0; rep < 2; ++rep) {
        const int d = tid + rep * 256;
        float acc = 0.f;
        for (int l = 0; l < LCTX; ++l) acc += s_sc[l] * context[(size_t)l * DD + d];
        cvec[d] = acc;
        out_ctx[(size_t)t * DD + d] = acc;
      }
    } else if (wg == 1) {
      // history attention over rows r < t  -> h_ctx
      float s = 0.f;
      for (int k = 0; k < DD; ++k) s += s_h[k] * Wha[(size_t)k * AA + tid];
      s_vecA[tid] = s;
      __syncthreads();
      if (t == 0) {
        hctx[tid] = 0.f; hctx[tid + 256] = 0.f;   // live==0 -> zeros
      } else {
        for (int rep = 0; rep < 2; ++rep) {
          const int r = tid + rep * 256;
          float sc = -3.0e38f;                     // masked (softmax weight 0)
          if (r < t) {
            sc = bh[0];
            for (int a = 0; a < AA; ++a)
              sc += wh[a] * tanhf(hwt[(size_t)a * T_STEPS + r] + bhh[a] + s_vecA[a]);
          }
          s_sc[r] = sc;
        }
        __syncthreads();
        s_red[tid] = fmaxf(s_sc[tid], s_sc[tid + 256]); __syncthreads();
        for (int st = 128; st > 0; st >>= 1) {
          if (tid < st) s_red[tid] = fmaxf(s_red[tid], s_red[tid + st]);
          __syncthreads();
        }
        const float mx = s_red[0]; __syncthreads();
        const float e0 = (s_sc[tid]       > -1e38f) ? __expf(s_sc[tid]       - mx) : 0.f;
        const float e1 = (s_sc[tid + 256] > -1e38f) ? __expf(s_sc[tid + 256] - mx) : 0.f;
        s_red[tid] = e0 + e1; __syncthreads();
        for (int st = 128; st > 0; st >>= 1) {
          if (tid < st) s_red[tid] += s_red[tid + st];
          __syncthreads();
        }
        const float inv = 1.f / s_red[0]; __syncthreads();
        s_sc[tid] = e0 * inv; s_sc[tid + 256] = e1 * inv; __syncthreads();
        for (int rep = 0; rep < 2; ++rep) {
          const int d = tid + rep * 256;
          float acc = 0.f;
          for (int r = 0; r < t; ++r) acc += s_sc[r] * out_h[(size_t)r * DD + d];
          hctx[d] = acc;
        }
      }
    } else {
      // gate slice: gates[c] = XWx[t,c] + h@Uh[:,c] + par_h@Pg[:,c]
      const int wgi   = wg - 2;                      // 0..29
      const int c0    = wgi * 68 + (wgi < 8 ? wgi : 8);
      const int ncols = 68 + (wgi < 8 ? 1 : 0);
      const int kh    = tid >> 7;                    // k-half: 0 or 1
      const int jc    = tid & 127;
      float g = 0.f;
      const int c = c0 + jc;
      if (jc < ncols) {
        const float* U  = Uh + (size_t)(kh * 256) * GG + c;
        const float* P  = Pg + (size_t)(kh * 256) * GG + c;
        const float* hh = s_h + kh * 256;
        const float* ph = out_h + (size_t)p * DD + kh * 256;  // parent hidden
        for (int k = 0; k < 256; ++k) {
          g += hh[k] * U[(size_t)k * GG];
          if (live) g += ph[k] * P[(size_t)k * GG];
        }
      }
      s_part[tid] = g; __syncthreads();
      if (kh == 0 && jc < ncols)
        gates[c] = s_part[tid] + s_part[tid + 128] + xwx[(size_t)t * GG + c];
    }
    tgt += NWG; global_barrier(barctr, tgt);

    // ---------------- PHASE B: gates[c] += cv@Cg + hc@Hg ---------------
    s_cv[tid] = cvec[tid]; s_cv[tid + 256] = cvec[tid + 256];
    s_hc[tid] = hctx[tid]; s_hc[tid + 256] = hctx[tid + 256];
    __syncthreads();
    {
      const int kq = tid >> 6;                       // 0..3
      const int jc = tid & 63;
      const int c  = wg * 64 + jc;
      const float* Cp = Cg + (size_t)(kq * 128) * GG + c;
      const float* Hp = Hg + (size_t)(kq * 128) * GG + c;
      float g = 0.f;
      for (int k = 0; k < 128; ++k) {
        const int ka = kq * 128 + k;
        g += s_cv[ka] * Cp[(size_t)k * GG] + s_hc[ka] * Hp[(size_t)k * GG];
      }
      s_part[tid] = g; __syncthreads();
      if (kq == 0)
        gates[c] += s_part[jc] + s_part[jc + 64] + s_part[jc + 128] + s_part[jc + 192];
    }
    tgt += NWG; global_barrier(barctr, tgt);

    // ---------------- PHASE C: h_new (redundant per WG, h stays in LDS)
    for (int rep = 0; rep < 2; ++rep) {
      const int d = tid + rep * 256;
      const float gi = gates[d], gf = gates[DD + d];
      const float gg = gates[2 * DD + d], go = gates[3 * DD + d];
      // faithful to the reference bug: carried cell state is always c0 == 0
      const float cn = sigmoidf_(gf) * 0.f + sigmoidf_(gi) * tanhf(gg);
      const float hn = sigmoidf_(go) * tanhf(cn);
      s_h[d] = hn;
      if ((d >> 4) == wg) out_h[(size_t)t * DD + d] = hn;   // owner writes
    }
    __syncthreads();

    // ---------------- PHASE D: append histWhh^T[:, t] = h_new @ Whh ----
    {
      const int al = tid & 7;                        // column within slice
      const int kq = tid >> 3;                       // 0..31 k-slices
      const int a  = wg * 8 + al;
      float s = 0.f;
      for (int k = kq * 16; k < kq * 16 + 16; ++k)
        s += s_h[k] * Whh[(size_t)k * AA + a];
      s_part[tid] = s; __syncthreads();
      if (tid < 8) {
        float tot = 0.f;
        for (int q = 0; q < 32; ++q) tot += s_part[q * 8 + tid];
        hwt[(size_t)(wg * 8 + tid) * T_STEPS + t] = tot;
      }
    }
    tgt += NWG; global_barrier(barctr, tgt);
  }
}

// ---------------------------------------------------------------------------
extern "C" void kernel_launch(void* const* d_in, const int* in_sizes, int n_in,
                              void* d_out, int out_size, void* d_ws, size_t ws_size,
                              hipStream_t stream) {
  const float* X       = (const float*)d_in[0];
  const float* context = (const float*)d_in[1];
  const float* h0      = (const float*)d_in[2];
  const float* Wx      = (const float*)d_in[3];
  const float* bx      = (const float*)d_in[4];
  const float* Uh      = (const float*)d_in[5];
  const float* Cg      = (const float*)d_in[6];
  const float* Pg      = (const float*)d_in[7];
  const float* Hg      = (const float*)d_in[8];
  const float* Wac     = (const float*)d_in[9];
  const float* bac     = (const float*)d_in[10];
  const float* Wah     = (const float*)d_in[11];
  const float* wa      = (const float*)d_in[12];
  const float* ba      = (const float*)d_in[13];
  const float* Whh     = (const float*)d_in[14];
  const float* bhh     = (const float*)d_in[15];
  const float* Wha     = (const float*)d_in[16];
  const float* wh      = (const float*)d_in[17];
  const float* bh      = (const float*)d_in[18];
  const int*   parent  = (const int*)d_in[19];

  float* out     = (float*)d_out;
  float* out_h   = out;                              // [T][D]
  float* out_ctx = out + (size_t)T_STEPS * DD;       // [T][D]
  float* ws      = (float*)d_ws;
  unsigned* barctr = (unsigned*)d_ws;

  // reset software-barrier counter (graph-capture safe)
  hipMemsetAsync(d_ws, 0, 256, stream);

  // Precompute XWx = X@Wx + bx   (512x512x2048, fp32 WMMA)
  wmma_gemm_f32_kernel<<<512, 256, 0, stream>>>(X, Wx, bx, ws + OFF_XWX,
                                                T_STEPS, GG, DD, 0);
  // Precompute ctxT = (context@Wac + bac)^T  stored [A][L] for coalesced scan
  wmma_gemm_f32_kernel<<<32, 256, 0, stream>>>(context, Wac, bac, ws + OFF_CTXT,
                                               LCTX, AA, DD, 1);
  // Persistent sequential scan over T=512 steps
  scan_kernel<<<NWG, NTHR, 0, stream>>>(h0, Uh, Cg, Pg, Hg, context,
                                        Wah, wa, ba, Whh, bhh, Wha, wh, bh,
                                        parent, out_h, out_ctx, ws, barctr);
}